// Attn_14078902796904
// MI455X (gfx1250) — compile-verified
//
#include <hip/hip_runtime.h>
#include <math.h>
#include <stdint.h>

#define SEQ 65536
#define HID 1024
#define WAVES_PER_BLOCK 8
#define ROWS_PER_WAVE 16
#define BLOCK_THREADS (WAVES_PER_BLOCK * 32)
#define ROWS_PER_BLOCK (WAVES_PER_BLOCK * ROWS_PER_WAVE)

// LDS slab geometry: 16 rows x 32 cols per wave, double buffered.
// Row stride 36 floats => ds_load_b128 banks (36r + 4j) mod 64 cover all 64
// banks exactly once across the 16 rows -> conflict-free fragment reads.
#define SLAB_COLS 32
#define LDS_ROW 36
#define SLAB_FLOATS (ROWS_PER_WAVE * LDS_ROW)     // 576 floats / buffer / wave
#define NUM_SLABS (HID / SLAB_COLS)               // 32
#define CHUNKS_PER_LANE 4                          // 128 x 16B chunks / 32 lanes

typedef float v2f __attribute__((ext_vector_type(2)));
typedef float v4f __attribute__((ext_vector_type(4)));
typedef float v8f __attribute__((ext_vector_type(8)));

// ---------------------------------------------------------------------------
// K1: energies[r] = dot(E[r,:], hidden) with V_WMMA_F32_16X16X4_F32.
// Global -> LDS via async-to-LDS (coalesced 16B/lane, ASYNCcnt), LDS -> WMMA
// A-fragments via broadcast ds_load_b128, B-fragments from scalar loads of
// hidden. One wave owns 16 sequence rows and its own double-buffered slab.
// ---------------------------------------------------------------------------
__global__ __launch_bounds__(BLOCK_THREADS) void energies_wmma(
    const float* __restrict__ hidden,
    const float* __restrict__ enc,
    float* __restrict__ energy)
{
    __shared__ __align__(16) float smem[WAVES_PER_BLOCK * 2 * SLAB_FLOATS];

    const int  lane    = threadIdx.x & 31;
    const int  wave    = threadIdx.x >> 5;
    const int  rowBase = (blockIdx.x * WAVES_PER_BLOCK + wave) * ROWS_PER_WAVE;
    const bool hi      = lane >= 16;
    const int  ldsWave = wave * 2 * SLAB_FLOATS;

    // Coalesced async stage of one 16x32 slab into LDS buffer b.
    auto stage = [&](int s, int b) {
        const float* g0 = enc + (size_t)rowBase * HID + s * SLAB_COLS;
        const int    l0 = ldsWave + b * SLAB_FLOATS;
#pragma unroll
        for (int i = 0; i < CHUNKS_PER_LANE; ++i) {
            const int t = (i << 5) + lane;        // chunk id 0..127
            const int r = t >> 3;                 // 8 x 16B chunks per row
            const int c = (t & 7) << 2;           // float offset in row
            const float* g = g0 + (size_t)r * HID + c;
            const uint32_t l =
                (uint32_t)(uintptr_t)&smem[l0 + r * LDS_ROW + c];
            asm volatile("global_load_async_to_lds_b128 %0, %1, off"
                         :: "v"(l), "v"(g) : "memory");
        }
    };

    stage(0, 0);
    stage(1, 1);

    v8f acc = {0.f, 0.f, 0.f, 0.f, 0.f, 0.f, 0.f, 0.f};
    const int rowOff = (lane & 15) * LDS_ROW;

    for (int s = 0; s < NUM_SLABS; ++s) {
        // Slab s resident once outstanding async ops <= (next slab's 4).
        if (s == NUM_SLABS - 1)
            asm volatile("s_wait_asynccnt 0x0" ::: "memory");
        else
            asm volatile("s_wait_asynccnt 0x4" ::: "memory");

        const int    base = ldsWave + (s & 1) * SLAB_FLOATS + rowOff;
        const float* hptr = hidden + s * SLAB_COLS;

#pragma unroll
        for (int j = 0; j < SLAB_COLS / 4; ++j) {
            // Lane L and L+16 read the same LDS address (broadcast): 16
            // rows x 16B, all 64 banks exactly once.
            v4f d  = *(const v4f*)&smem[base + 4 * j];
            v4f hv = *(const v4f*)(hptr + 4 * j);   // uniform -> s_load_b128
            v2f a, b;
            a.x = hi ? d[2]  : d[0];
            a.y = hi ? d[3]  : d[1];
            b.x = hi ? hv[2] : hv[0];
            b.y = hi ? hv[3] : hv[1];
            acc = __builtin_amdgcn_wmma_f32_16x16x4_f32(
                false, a, false, b, (short)0, acc, false, false);
        }

        if (s + 2 < NUM_SLABS) {
            // Fragment reads of the buffer being overwritten must be done.
            asm volatile("s_wait_dscnt 0x0" ::: "memory");
            stage(s + 2, s & 1);
        }
    }

    // D: VGPR j -> row j (lanes 0..15) / row 8+j (lanes 16..31); cols identical.
    if ((lane & 15) == 0) {
        const int r = rowBase + (lane >> 4) * 8;
#pragma unroll
        for (int j = 0; j < 8; ++j) energy[r + j] = acc[j];
    }
}

// ---------------------------------------------------------------------------
// K2: single-block global max + sum(exp(e - max)) over SEQ elements.
// ---------------------------------------------------------------------------
__global__ __launch_bounds__(1024) void softmax_reduce(
    const float* __restrict__ e, float* __restrict__ ms)
{
    __shared__ float red[1024];
    const int tid = threadIdx.x;

    float mx = -INFINITY;
    for (int i = tid; i < SEQ; i += 1024) mx = fmaxf(mx, e[i]);
    red[tid] = mx;
    __syncthreads();
    for (int s = 512; s > 0; s >>= 1) {
        if (tid < s) red[tid] = fmaxf(red[tid], red[tid + s]);
        __syncthreads();
    }
    const float gmax = red[0];
    __syncthreads();

    float sum = 0.f;
    for (int i = tid; i < SEQ; i += 1024) sum += __expf(e[i] - gmax);
    red[tid] = sum;
    __syncthreads();
    for (int s = 512; s > 0; s >>= 1) {
        if (tid < s) red[tid] += red[tid + s];
        __syncthreads();
    }
    if (tid == 0) { ms[0] = gmax; ms[1] = red[0]; }
}

// ---------------------------------------------------------------------------
// K3: out[i] = exp(out[i] - max) / sum    (in place, one element per thread)
// ---------------------------------------------------------------------------
__global__ __launch_bounds__(256) void softmax_norm(
    float* __restrict__ out, const float* __restrict__ ms)
{
    const int i = blockIdx.x * 256 + threadIdx.x;
    const float m   = ms[0];
    const float inv = 1.f / ms[1];
    out[i] = __expf(out[i] - m) * inv;
}

extern "C" void kernel_launch(void* const* d_in, const int* in_sizes, int n_in,
                              void* d_out, int out_size, void* d_ws, size_t ws_size,
                              hipStream_t stream)
{
    const float* hidden = (const float*)d_in[0];   // [1024]
    const float* enc    = (const float*)d_in[1];   // [65536, 1024]
    float*       out    = (float*)d_out;           // [65536] (== [1,1,65536])
    float*       ms     = (float*)d_ws;            // 2 floats: {max, sum}

    const int grid1 = SEQ / ROWS_PER_BLOCK;        // 512 blocks of 8 waves
    hipLaunchKernelGGL(energies_wmma, dim3(grid1), dim3(BLOCK_THREADS), 0, stream,
                       hidden, enc, out);
    hipLaunchKernelGGL(softmax_reduce, dim3(1), dim3(1024), 0, stream, out, ms);
    hipLaunchKernelGGL(softmax_norm, dim3(SEQ / 256), dim3(256), 0, stream, out, ms);
}